// FixedStackRNNG_89094801588644
// MI455X (gfx1250) — compile-verified
//
#include <hip/hip_runtime.h>
#include <hip/hip_bf16.h>
#include <stdint.h>

// Problem shape (fixed by the reference)
#define B_ 32
#define K_ 128
#define S_ 48
#define L_ 2
#define H_ 128
#define A_ 500

#define HID_ROW (S_*L_*H_)   // 12288 floats = 48KB per beam row
#define TREE_ROW (S_*H_)     // 6144 floats  = 24KB per beam row

// Flat float offsets of each tuple member inside d_out (return order)
#define O_GEN   0
#define O_DISC  (B_*K_)
#define O_HID   (2*B_*K_)
#define O_CELL  (O_HID  + B_*K_*HID_ROW)
#define O_TREE  (O_CELL + B_*K_*HID_ROW)
#define O_WID   (O_TREE + B_*K_*TREE_ROW)
#define O_NCONS (O_WID  + B_)
#define O_NOPEN (O_NCONS + B_*K_)
#define O_ACT   (O_NOPEN + B_*K_)
#define O_APOS  (O_ACT  + B_*K_*A_)
#define O_PTR   (O_APOS + B_*K_)

// Streaming data: used exactly once -> keep it out of L2 residency.
#define NT_LD  " th:TH_LOAD_NT"
#define NT_ST  " th:TH_STORE_NT"

// ---------------------------------------------------------------------------
// Phase 1: per-batch mask + stable descending rank-sort (K=128), and emit all
// the small outputs plus the beam permutation into workspace.
// Stable: ties broken by original index ascending == jnp.argsort(-masked).
// ---------------------------------------------------------------------------
__global__ __launch_bounds__(K_) void sort_small_kernel(
    const float* __restrict__ gen, const float* __restrict__ disc,
    const int* __restrict__ widths, const int* __restrict__ ncons,
    const int* __restrict__ nopen, const int* __restrict__ apos,
    const int* __restrict__ ptr, const int* __restrict__ size,
    float* __restrict__ out, int* __restrict__ srt)
{
    __shared__ float sval[K_];
    __shared__ int   sidx[K_];
    const int b = blockIdx.x;
    const int i = threadIdx.x;
    const int base = b * K_;

    const int w = widths[b];
    const float v = (i >= w) ? -1e30f : gen[base + i];
    sval[i] = v;
    __syncthreads();

    int rank = 0;
#pragma unroll 8
    for (int j = 0; j < K_; ++j) {
        const float o = sval[j];
        rank += (o > v) || ((o == v) && (j < i));
    }
    sidx[rank] = i;                 // ranks are unique -> no conflicts
    __syncthreads();

    const int src = sidx[i];        // source beam feeding sorted slot i
    out[O_GEN   + base + i] = sval[src];
    out[O_DISC  + base + i] = disc[base + src];
    out[O_NCONS + base + i] = (float)ncons[base + src];
    out[O_NOPEN + base + i] = (float)nopen[base + src];
    out[O_APOS  + base + i] = (float)apos[base + src];
    out[O_PTR   + base + i] = (float)ptr[base + src];
    srt[base + i] = src;

    if (i == 0) {
        const int sz = size[0];
        out[O_WID + b] = (float)((w < sz) ? w : sz);
    }
}

// ---------------------------------------------------------------------------
// Phase 2: bulk row permutation via the CDNA5 async global<->LDS DMA path.
// One 256-thread block per destination row, 16KB LDS staging. Main rounds
// move 4096 floats (16KB): 4x GLOBAL_LOAD_ASYNC_TO_LDS_B128 per lane,
// s_wait_asynccnt 0, 4x GLOBAL_STORE_ASYNC_FROM_LDS_B128, wait before LDS
// reuse. A 2048-float (8KB) tail round covers the trees row (6144 = 4096 +
// 2048). Per the ISA, INST_OFFSET is added to BOTH the LDS and global
// address, so one LDS-base VGPR + offset:{4096,8192,12288} addresses all
// four slots. ASYNCcnt retires loads vs stores out of order, so each round
// is bracketed by full waits; wave-level parallelism (4096 blocks x 8 waves
// per launch) hides the per-wave serialization against the HBM roofline.
// ---------------------------------------------------------------------------
__global__ __launch_bounds__(256) void gather_rows_async_kernel(
    const float* __restrict__ src, float* __restrict__ dst,
    const int* __restrict__ srt, int rowLen)
{
    __shared__ float4 buf[1024];    // 16KB staging: four 4KB slots

    const int r  = blockIdx.x;      // destination row in [0, B*K)
    const int b  = r >> 7;          // / K_
    const int sb = srt[r];          // source beam

    const float* sp0 = src + (size_t)((b << 7) + sb) * (size_t)rowLen;
    float*       dp0 = dst + (size_t)r * (size_t)rowLen;

    // LDS byte address of this lane's slot (flat shared ptr truncates to LDS addr)
    const unsigned lds = (unsigned)(uintptr_t)&buf[threadIdx.x];
    unsigned long long sp = (unsigned long long)(uintptr_t)sp0 + (unsigned long long)threadIdx.x * 16ull;
    unsigned long long dp = (unsigned long long)(uintptr_t)dp0 + (unsigned long long)threadIdx.x * 16ull;

    int rem = rowLen;
    while (rem >= 4096) {           // 16KB per round
        asm volatile("global_load_async_to_lds_b128 %0, %1, off" NT_LD
                     :: "v"(lds), "v"(sp) : "memory");
        asm volatile("global_load_async_to_lds_b128 %0, %1, off offset:4096" NT_LD
                     :: "v"(lds), "v"(sp) : "memory");
        asm volatile("global_load_async_to_lds_b128 %0, %1, off offset:8192" NT_LD
                     :: "v"(lds), "v"(sp) : "memory");
        asm volatile("global_load_async_to_lds_b128 %0, %1, off offset:12288" NT_LD
                     :: "v"(lds), "v"(sp) : "memory");
        asm volatile("s_wait_asynccnt 0" ::: "memory");
        asm volatile("global_store_async_from_lds_b128 %0, %1, off" NT_ST
                     :: "v"(dp), "v"(lds) : "memory");
        asm volatile("global_store_async_from_lds_b128 %0, %1, off offset:4096" NT_ST
                     :: "v"(dp), "v"(lds) : "memory");
        asm volatile("global_store_async_from_lds_b128 %0, %1, off offset:8192" NT_ST
                     :: "v"(dp), "v"(lds) : "memory");
        asm volatile("global_store_async_from_lds_b128 %0, %1, off offset:12288" NT_ST
                     :: "v"(dp), "v"(lds) : "memory");
        asm volatile("s_wait_asynccnt 0" ::: "memory");  // LDS reusable next round
        sp += 16384ull;
        dp += 16384ull;
        rem -= 4096;
    }
    if (rem >= 2048) {              // 8KB tail (trees rows)
        asm volatile("global_load_async_to_lds_b128 %0, %1, off" NT_LD
                     :: "v"(lds), "v"(sp) : "memory");
        asm volatile("global_load_async_to_lds_b128 %0, %1, off offset:4096" NT_LD
                     :: "v"(lds), "v"(sp) : "memory");
        asm volatile("s_wait_asynccnt 0" ::: "memory");
        asm volatile("global_store_async_from_lds_b128 %0, %1, off" NT_ST
                     :: "v"(dp), "v"(lds) : "memory");
        asm volatile("global_store_async_from_lds_b128 %0, %1, off offset:4096" NT_ST
                     :: "v"(dp), "v"(lds) : "memory");
        asm volatile("s_wait_asynccnt 0" ::: "memory");
    }
}

// ---------------------------------------------------------------------------
// Phase 3: actions gather (int -> float conversion forces the VGPR path).
// ---------------------------------------------------------------------------
__global__ __launch_bounds__(256) void gather_actions_kernel(
    const int* __restrict__ act, const int* __restrict__ srt,
    float* __restrict__ out)
{
    const int e = blockIdx.x * blockDim.x + threadIdx.x;
    if (e >= B_ * K_ * A_) return;
    const int bk = e / A_;
    const int a  = e - bk * A_;
    const int b  = bk >> 7;
    const int src = srt[bk];
    out[O_ACT + e] = (float)act[(size_t)((b << 7) + src) * A_ + a];
}

// ---------------------------------------------------------------------------
extern "C" void kernel_launch(void* const* d_in, const int* in_sizes, int n_in,
                              void* d_out, int out_size, void* d_ws, size_t ws_size,
                              hipStream_t stream)
{
    (void)in_sizes; (void)n_in; (void)out_size; (void)ws_size;

    const float* gen    = (const float*)d_in[0];
    const float* disc   = (const float*)d_in[1];
    const float* hidd   = (const float*)d_in[2];
    const float* cells  = (const float*)d_in[3];
    const float* trees  = (const float*)d_in[4];
    const int*   widths = (const int*)d_in[5];
    const int*   ncons  = (const int*)d_in[6];
    const int*   nopen  = (const int*)d_in[7];
    const int*   act    = (const int*)d_in[8];
    const int*   apos   = (const int*)d_in[9];
    const int*   ptr    = (const int*)d_in[10];
    const int*   size   = (const int*)d_in[11];

    float* out = (float*)d_out;
    int*   srt = (int*)d_ws;        // B*K ints of scratch (16KB)

    sort_small_kernel<<<B_, K_, 0, stream>>>(gen, disc, widths, ncons, nopen,
                                             apos, ptr, size, out, srt);

    gather_rows_async_kernel<<<B_ * K_, 256, 0, stream>>>(hidd,  out + O_HID,  srt, HID_ROW);
    gather_rows_async_kernel<<<B_ * K_, 256, 0, stream>>>(cells, out + O_CELL, srt, HID_ROW);
    gather_rows_async_kernel<<<B_ * K_, 256, 0, stream>>>(trees, out + O_TREE, srt, TREE_ROW);

    const int nAct = B_ * K_ * A_;
    gather_actions_kernel<<<(nAct + 255) / 256, 256, 0, stream>>>(act, srt, out);
}